// SNN_MMSE_52639119180422
// MI455X (gfx1250) — compile-verified
//
#include <hip/hip_runtime.h>

typedef float v2f __attribute__((ext_vector_type(2)));
typedef float v4f __attribute__((ext_vector_type(4)));
typedef float v8f __attribute__((ext_vector_type(8)));

#define T_STEPS 128
#define BATCH   2048
#define IN      64
#define HID     256
#define OUT     16
#define DT_TAU  0.16666666666666666f
#define DECAY   (1.0f - DT_TAU)

#define NWAVES    8    // waves per workgroup; wave owns 32 H-columns (2 N-tiles)
#define ZS_STRIDE 260  // padded row stride for spike tile in LDS

static __device__ __forceinline__ v8f splat8(float v) {
    v8f r;
#pragma unroll
    for (int i = 0; i < 8; ++i) r[i] = v;
    return r;
}

__global__ __launch_bounds__(32 * NWAVES) void snn_scan_kernel(
    const float* __restrict__ x,    // [T, B, I]
    const float* __restrict__ w1,   // [H, I]
    const float* __restrict__ w2,   // [O, H]
    float* __restrict__ out,        // [B, O] (+1 scalar written by finalize)
    unsigned* __restrict__ spike_cnt)
{
    __shared__ float zs[16 * ZS_STRIDE];           // spike tile [m][h]; column slices are wave-private
    __shared__ float g2p[2][NWAVES][16 * 16];      // double-buffered layer-2 partials
    __shared__ unsigned wg_cnt;

    const int tid  = threadIdx.x;
    const int lane = tid & 31;
    const int wv   = tid >> 5;        // wave 0..7 -> H slice [32*wv, 32*wv+32)
    const int row  = lane & 15;       // A-layout M index / B,C,D-layout N index
    const int hi   = lane >> 4;       // half-wave select
    const int cb   = 2 * hi;          // K sub-offset within a 4-wide K slice
    const int b0   = blockIdx.x * 16; // batch tile base

    if (tid == 0) wg_cnt = 0u;

    // Loop-invariant w1 B-fragments in registers: B[k][n] = w1[n*IN + k]
    v2f b1[2][16];
#pragma unroll
    for (int j = 0; j < 2; ++j)
#pragma unroll
        for (int kk = 0; kk < 16; ++kk)
            b1[j][kk] = *(const v2f*)&w1[(32 * wv + 16 * j + row) * IN + 4 * kk + cb];

    // w2 B-fragments for this wave's K slice. B[k][n] = w2[n*HID + k]
    v2f b2[8];
#pragma unroll
    for (int kk = 0; kk < 8; ++kk)
        b2[kk] = *(const v2f*)&w2[row * HID + 32 * wv + 4 * kk + cb];

    // Recurrent state in registers (C/D fragment layout)
    v8f v1[2], i1[2];
#pragma unroll
    for (int j = 0; j < 2; ++j) { v1[j] = splat8(0.f); i1[j] = splat8(0.f); }
    v8f v2s  = splat8(0.f);                 // live only in wave 0
    v8f i2s  = splat8(0.f);
    v8f ymax = splat8(-3.402823466e38f);
    float fcnt = 0.0f;                      // exact integer-valued accumulator

    __syncthreads();

    for (int t = 0; t < T_STEPS; ++t) {
        // Prefetch next step's contiguous 4 KB x tile into WGP$/L2
        if (wv == 0 && t + 1 < T_STEPS)
            __builtin_prefetch(x + ((size_t)(t + 1) * BATCH + b0) * IN + lane * 32, 0, 3);

        // ---- A-fragments of x tile: A[m=row][k], k = 4*kk + cb + {0,1}
        const float* xt = x + ((size_t)t * BATCH + b0 + row) * IN;
        v2f a[16];
#pragma unroll
        for (int kk = 0; kk < 16; ++kk)
            a[kk] = *(const v2f*)&xt[4 * kk + cb];

        // ---- Layer 1: LIF update + g1 via WMMA chain (C seeded with i1*decay)
#pragma unroll
        for (int j = 0; j < 2; ++j) {
            const int hbase = 32 * wv + 16 * j;
#pragma unroll
            for (int r = 0; r < 8; ++r) {
                float vd = v1[j][r] + DT_TAU * (i1[j][r] - v1[j][r]);
                const bool s = vd > 1.0f;             // heaviside(v_dec - 1)
                float zr = s ? 1.0f : 0.0f;
                v1[j][r] = s ? 0.0f : vd;             // reset where spiked
                fcnt += zr;                           // exact float count
                zs[(r + 8 * hi) * ZS_STRIDE + hbase + row] = zr;
            }
            v8f acc = i1[j] * DECAY;                  // synaptic decay
#pragma unroll
            for (int kk = 0; kk < 16; ++kk) {
                acc = __builtin_amdgcn_wmma_f32_16x16x4_f32(
                        false, a[kk], false, b1[j][kk], (short)0, acc, false, false);
            }
            i1[j] = acc;                              // i1*decay + g1
        }

        // zs column slice is wave-private (written and read by the same wave);
        // same-wave LDS ops are in-order in HW, only stop compiler reordering.
        asm volatile("" ::: "memory");

        // ---- Layer 2: load all z A-fragments first (batch the ds loads into
        // one dscnt wait), then run the 8-deep WMMA chain.
        v2f a2[8];
#pragma unroll
        for (int kk = 0; kk < 8; ++kk)
            a2[kk] = *(const v2f*)&zs[row * ZS_STRIDE + 32 * wv + 4 * kk + cb];

        v8f acc2 = splat8(0.f);
#pragma unroll
        for (int kk = 0; kk < 8; ++kk)
            acc2 = __builtin_amdgcn_wmma_f32_16x16x4_f32(
                        false, a2[kk], false, b2[kk], (short)0, acc2, false, false);

        float* gp = &g2p[t & 1][wv][0];
#pragma unroll
        for (int r = 0; r < 8; ++r)
            gp[(r + 8 * hi) * 16 + row] = acc2[r];

        // Single split-barrier per step: publishes parity-t partials to wave 0.
        __syncthreads();

        // ---- LI readout update: only wave 0 owns v2/i2/ymax
        if (wv == 0) {
            const float* gr = &g2p[t & 1][0][0];
#pragma unroll
            for (int r = 0; r < 8; ++r) {
                const int idx = (r + 8 * hi) * 16 + row;
                float g2 = 0.0f;
#pragma unroll
                for (int w = 0; w < NWAVES; ++w) g2 += gr[w * 256 + idx];
                float v2n = v2s[r] + DT_TAU * (i2s[r] - v2s[r]);
                i2s[r] = i2s[r] * DECAY + g2;
                v2s[r] = v2n;
                ymax[r] = fmaxf(ymax[r], v2n);
            }
        }
    }

    // ---- score = max_t v2 : D-layout element (m = r+8*hi, o = row)
    if (wv == 0) {
#pragma unroll
        for (int r = 0; r < 8; ++r)
            out[(size_t)(b0 + r + 8 * hi) * OUT + row] = ymax[r];
    }

    atomicAdd(&wg_cnt, (unsigned)fcnt);
    __syncthreads();
    if (tid == 0) atomicAdd(spike_cnt, wg_cnt);
}

__global__ void snn_finalize_kernel(float* __restrict__ out,
                                    const unsigned* __restrict__ spike_cnt)
{
    out[BATCH * OUT] = (float)(*spike_cnt) * (1.0f / (float)BATCH);
}

extern "C" void kernel_launch(void* const* d_in, const int* in_sizes, int n_in,
                              void* d_out, int out_size, void* d_ws, size_t ws_size,
                              hipStream_t stream)
{
    const float* x  = (const float*)d_in[0];
    const float* w1 = (const float*)d_in[1];
    const float* w2 = (const float*)d_in[2];
    float* out = (float*)d_out;
    unsigned* cnt = (unsigned*)d_ws;

    hipMemsetAsync(d_ws, 0, sizeof(unsigned), stream);
    snn_scan_kernel<<<dim3(BATCH / 16), dim3(32 * NWAVES), 0, stream>>>(x, w1, w2, out, cnt);
    snn_finalize_kernel<<<1, 1, 0, stream>>>(out, cnt);
}